// ScalarDotProductCriticNetworkV8_78993038508708
// MI455X (gfx1250) — compile-verified
//
#include <hip/hip_runtime.h>
#include <math.h>

// Problem dims (fixed by reference)
#define BB  512
#define NN  16
#define AA  16
#define OBSD 112
#define DD  128
#define HH  64
#define FIN 128

typedef __attribute__((ext_vector_type(2))) float v2f;
typedef __attribute__((ext_vector_type(8))) float v8f;

static __device__ __forceinline__ v8f wmma_f32(v2f a, v2f b, v8f c) {
  // D = A(16x4,f32) * B(4x16,f32) + C(16x16,f32)
  return __builtin_amdgcn_wmma_f32_16x16x4_f32(false, a, false, b, (short)0, c,
                                               false, false);
}

#define LSTRIDE 132  // 128 + 4 pad -> row advances bank by 4 (64 banks)
#define HSTRIDE 66   // 64  + 2 pad

__global__ __launch_bounds__(256) void critic_kernel(
    const float* __restrict__ states,    // [B,N,OBS]
    const float* __restrict__ policies,  // [B,N,A]
    const float* __restrict__ actions,   // [B,N,A]
    const float* __restrict__ Wk, const float* __restrict__ bk,
    const float* __restrict__ Wq, const float* __restrict__ bq,
    const float* __restrict__ Wv, const float* __restrict__ bv,
    const float* __restrict__ W1, const float* __restrict__ W2,
    float* __restrict__ out_value,       // [B,N,N]
    float* __restrict__ out_weight)      // [B,N,N]
{
  __shared__ float s_inA[NN][LSTRIDE];  // obs||actions ; reused as S after phase C
  __shared__ float s_inP[NN][LSTRIDE];  // obs||policies
  __shared__ float s_k  [NN][LSTRIDE];
  __shared__ float s_q  [NN][LSTRIDE];
  __shared__ float s_avA[NN][LSTRIDE];  // tanh(obs_act @ Wv^T + bv)
  __shared__ float s_dlt[NN][LSTRIDE];  // av_pol - av_act
  __shared__ float s_SW [NN][HSTRIDE];  // S @ W1^T
  __shared__ float s_DW [NN][HSTRIDE];  // delta @ W1^T
  __shared__ float s_sc [NN][NN];       // raw scores (atomic-accumulated)
  __shared__ float s_w  [NN][NN];       // softmax weights
  __shared__ float s_W2 [HH];

  const int t    = threadIdx.x;
  const int b    = blockIdx.x;
  const int lane = t & 31;
  const int wv   = t >> 5;        // wave 0..7 -> d-slice
  const int ln   = lane & 15;     // A-row / B-col / C-col index
  const int hi   = lane >> 4;     // half-wave select
  const int kk   = 2 * hi;        // K offset of vgpr pair within a 4-step
  const int d0   = wv * 16;

  // ---------------- Phase 0: stage inputs, zero score accum ----------------
  for (int idx = t; idx < NN * FIN; idx += 256) {
    int i = idx >> 7, c = idx & 127;
    if (c < OBSD) {
      float sv = states[(b * NN + i) * OBSD + c];
      s_inA[i][c] = sv;
      s_inP[i][c] = sv;
    } else {
      s_inA[i][c] = actions [(b * NN + i) * AA + (c - OBSD)];
      s_inP[i][c] = policies[(b * NN + i) * AA + (c - OBSD)];
    }
  }
  if (t < NN * NN) (&s_sc[0][0])[t] = 0.0f;
  if (t < HH) s_W2[t] = W2[t];
  __syncthreads();

  // ------------- Phase A1: k,q = states @ {Wk,Wq}^T + bias (K=112) ----------
  {
    v8f ck = {}; v8f cq = {};
    const float* wkrow = Wk + (d0 + ln) * OBSD;
    const float* wqrow = Wq + (d0 + ln) * OBSD;
    for (int f = 0; f < OBSD; f += 4) {
      v2f a;  a.x  = s_inA[ln][f + kk]; a.y  = s_inA[ln][f + kk + 1];
      v2f bk_; bk_.x = wkrow[f + kk];   bk_.y = wkrow[f + kk + 1];
      v2f bq_; bq_.x = wqrow[f + kk];   bq_.y = wqrow[f + kk + 1];
      ck = wmma_f32(a, bk_, ck);        // shared A operand, 2 WMMAs / step
      cq = wmma_f32(a, bq_, cq);
    }
    float bbk = bk[d0 + ln], bbq = bq[d0 + ln];
#pragma unroll
    for (int r = 0; r < 8; ++r) {
      s_k[r + 8 * hi][d0 + ln] = ck[r] + bbk;
      s_q[r + 8 * hi][d0 + ln] = cq[r] + bbq;
    }
  }

  // ------ Phase A2: av_act/av_pol = tanh(in @ Wv^T + bv) (K=128, fused) -----
  {
    v8f ca = {}; v8f cp = {};
    const float* wvrow = Wv + (d0 + ln) * FIN;
    for (int f = 0; f < FIN; f += 4) {
      v2f aa; aa.x = s_inA[ln][f + kk]; aa.y = s_inA[ln][f + kk + 1];
      v2f ap; ap.x = s_inP[ln][f + kk]; ap.y = s_inP[ln][f + kk + 1];
      v2f bb; bb.x = wvrow[f + kk];     bb.y = wvrow[f + kk + 1];
      ca = wmma_f32(aa, bb, ca);        // shared B (Wv) operand
      cp = wmma_f32(ap, bb, cp);
    }
    float bbv = bv[d0 + ln];
#pragma unroll
    for (int r = 0; r < 8; ++r) {
      float va = tanhf(ca[r] + bbv);
      float vp = tanhf(cp[r] + bbv);
      s_avA[r + 8 * hi][d0 + ln] = va;
      s_dlt[r + 8 * hi][d0 + ln] = vp - va;
    }
  }
  __syncthreads();

  // --- Phase B: scores[j,i] += (1/sqrt(128)) * sum_{d in slice} k[j,d]q[i,d]
  {
    v8f cs = {};
    for (int k0 = 0; k0 < 16; k0 += 4) {
      int col = d0 + k0 + kk;
      v2f a;   a.x   = s_k[ln][col]; a.y   = s_k[ln][col + 1];
      v2f bq2; bq2.x = s_q[ln][col]; bq2.y = s_q[ln][col + 1];
      cs = wmma_f32(a, bq2, cs);
    }
    const float scl = 0.08838834764831845f;  // 1/sqrt(128)
#pragma unroll
    for (int r = 0; r < 8; ++r)
      atomicAdd(&s_sc[r + 8 * hi][ln], cs[r] * scl);  // ds_add_f32 cross-wave reduce
  }
  __syncthreads();

  // ------------- Phase C: softmax over i per key row j; emit weight ---------
  if (t < NN) {
    int j = t;
    float m = -3.4e38f;
#pragma unroll
    for (int i = 0; i < NN; ++i) m = fmaxf(m, s_sc[j][i]);
    float e[NN]; float sum = 0.f;
#pragma unroll
    for (int i = 0; i < NN; ++i) { e[i] = expf(s_sc[j][i] - m); sum += e[i]; }
    float inv = 1.f / sum;
#pragma unroll
    for (int i = 0; i < NN; ++i) {
      float w = e[i] * inv;
      s_w[j][i] = w;
      out_weight[(b * NN + j) * NN + i] = w;
    }
  }
  __syncthreads();

  // ----- Phase D: S[a,d] = sum_i w[a,i]*av_act[i,d] (K=16); store in s_inA --
  {
    v8f c = {};
    for (int k0 = 0; k0 < NN; k0 += 4) {
      v2f a;  a.x  = s_w  [ln][k0 + kk];          a.y  = s_w  [ln][k0 + kk + 1];
      v2f bb; bb.x = s_avA[k0 + kk][d0 + ln];     bb.y = s_avA[k0 + kk + 1][d0 + ln];
      c = wmma_f32(a, bb, c);
    }
#pragma unroll
    for (int r = 0; r < 8; ++r)
      s_inA[r + 8 * hi][d0 + ln] = c[r];  // s_inA now holds S
  }
  __syncthreads();

  // --- Phase E: SW = S @ W1^T (waves 0-3), DW = delta @ W1^T (waves 4-7) ----
  {
    const float* src = (wv < 4) ? &s_inA[0][0] : &s_dlt[0][0];
    float (*dst)[HSTRIDE] = (wv < 4) ? s_SW : s_DW;
    const int h0 = (wv & 3) * 16;
    const float* w1row = W1 + (h0 + ln) * DD;
    v8f c = {};
    for (int f = 0; f < DD; f += 4) {
      v2f a;  a.x  = src[ln * LSTRIDE + f + kk]; a.y  = src[ln * LSTRIDE + f + kk + 1];
      v2f bb; bb.x = w1row[f + kk];              bb.y = w1row[f + kk + 1];
      c = wmma_f32(a, bb, c);
    }
#pragma unroll
    for (int r = 0; r < 8; ++r)
      dst[r + 8 * hi][h0 + ln] = c[r];
  }
  __syncthreads();

  // -- Phase F: value[a,j] = sum_h lrelu((SW[a,h]+w[a,j]DW[j,h])/16) * W2[h] -
  {
    int a = t >> 4, j = t & 15;
    float w = s_w[a][j];
    float acc = 0.f;
#pragma unroll 8
    for (int h = 0; h < HH; ++h) {
      float x = (s_SW[a][h] + w * s_DW[j][h]) * (1.0f / 16.0f);
      x = (x > 0.f) ? x : 0.01f * x;   // leaky_relu(0.01)
      acc += x * s_W2[h];
    }
    out_value[(b * NN + a) * NN + j] = acc;
  }
}

extern "C" void kernel_launch(void* const* d_in, const int* in_sizes, int n_in,
                              void* d_out, int out_size, void* d_ws, size_t ws_size,
                              hipStream_t stream) {
  const float* states   = (const float*)d_in[0];
  const float* policies = (const float*)d_in[1];
  const float* actions  = (const float*)d_in[2];
  const float* Wk       = (const float*)d_in[3];
  const float* bk       = (const float*)d_in[4];
  const float* Wq       = (const float*)d_in[5];
  const float* bq       = (const float*)d_in[6];
  const float* Wv       = (const float*)d_in[7];
  const float* bv       = (const float*)d_in[8];
  const float* W1       = (const float*)d_in[9];
  const float* W2       = (const float*)d_in[10];

  float* out_value  = (float*)d_out;               // [B,N,N,1] flat
  float* out_weight = out_value + BB * NN * NN;    // [B,N,N] flat

  critic_kernel<<<BB, 256, 0, stream>>>(states, policies, actions,
                                        Wk, bk, Wq, bq, Wv, bv, W1, W2,
                                        out_value, out_weight);
}